// MaxUnpooling2D_77902116814930
// MI455X (gfx1250) — compile-verified
//
#include <hip/hip_runtime.h>

// ---------------------------------------------------------------------------
// MaxUnpooling2D (B=32, H=112, W=112, C=64, upscale 2x2), fp32, scatter-add.
//   out = zeros([B, 224, 224, 64]);  out[b, m/(224*64), (m/64)%224, c] += upd
// Pure memory-bound scatter: no matmul content -> no WMMA path applies.
// Strategy: B128 zero-fill (RT, keep lines in 192MB L2 for the atomics),
// NT-hint B128 streaming loads of updates/mask, hardware no-return
// global_atomic_add_f32 for the duplicate-summing scatter.
// ---------------------------------------------------------------------------

typedef float v4f __attribute__((ext_vector_type(4)));
typedef int   v4i __attribute__((ext_vector_type(4)));

namespace {
constexpr unsigned B_    = 32;
constexpr unsigned H_    = 112;
constexpr unsigned W_    = 112;
constexpr unsigned C_    = 64;
constexpr unsigned OUTH  = H_ * 2;                 // 224
constexpr unsigned OUTW  = W_ * 2;                 // 224
constexpr unsigned OROW  = OUTW * C_;              // 14336 = stride of one output row
constexpr unsigned OIMG  = OUTH * OROW;            // 3,211,264 per batch image
constexpr unsigned IIMG  = H_ * W_ * C_;           // 802,816 per batch image
constexpr unsigned N_IN  = B_ * IIMG;              // 25,690,112 input elements
constexpr unsigned N_OUT = B_ * OIMG;              // 102,760,448 output elements
static_assert((N_IN % 4u) == 0u, "vector-4 path requires N_IN % 4 == 0");
static_assert((N_OUT % 4u) == 0u, "vector-4 path requires N_OUT % 4 == 0");
} // namespace

// --- Kernel 1: zero the output with 16B stores (global_store_b128) ---------
__global__ __launch_bounds__(256) void unpool_zero_kernel(v4f* __restrict__ out,
                                                          unsigned n4) {
    unsigned i = blockIdx.x * 256u + threadIdx.x;
    if (i < n4) {
        v4f z = (v4f)0.0f;
        out[i] = z;   // default RT temporal hint: leave zeroed lines L2-resident
    }
}

// --- Kernel 2: decode mask + scatter-add via global_atomic_add_f32 ---------
__global__ __launch_bounds__(256) void unpool_scatter_kernel(
        const v4f* __restrict__ upd4,
        const v4i* __restrict__ msk4,
        float* __restrict__ out,
        unsigned n4) {
    unsigned t = blockIdx.x * 256u + threadIdx.x;
    if (t >= n4) return;

    // Streaming inputs: non-temporal loads (TH=NT) so the 205MB of one-touch
    // input does not evict the scatter target from the 192MB L2.
    v4f u = __builtin_nontemporal_load(&upd4[t]);
    v4i m = __builtin_nontemporal_load(&msk4[t]);

    unsigned i4   = t * 4u;            // flat element index of lane's 1st elem
    unsigned b    = i4 / IIMG;         // batch (shared by the 4 elements)
    unsigned c0   = i4 & (C_ - 1u);    // channel of 1st elem (4 | C, so same b,h,w)
    unsigned base = b * OIMG + c0;     // output base for this quad

    // Per element j: y = m/(OUTW*C); x = (m/C) % OUTW; dst = base + j + y*OROW + x*C
    {
        unsigned mm = (unsigned)m.x;
        unsigned y  = mm / OROW;
        unsigned x  = (mm >> 6) % OUTW;
        unsafeAtomicAdd(out + (base + 0u + y * OROW + x * C_), u.x);
    }
    {
        unsigned mm = (unsigned)m.y;
        unsigned y  = mm / OROW;
        unsigned x  = (mm >> 6) % OUTW;
        unsafeAtomicAdd(out + (base + 1u + y * OROW + x * C_), u.y);
    }
    {
        unsigned mm = (unsigned)m.z;
        unsigned y  = mm / OROW;
        unsigned x  = (mm >> 6) % OUTW;
        unsafeAtomicAdd(out + (base + 2u + y * OROW + x * C_), u.z);
    }
    {
        unsigned mm = (unsigned)m.w;
        unsigned y  = mm / OROW;
        unsigned x  = (mm >> 6) % OUTW;
        unsafeAtomicAdd(out + (base + 3u + y * OROW + x * C_), u.w);
    }
}

extern "C" void kernel_launch(void* const* d_in, const int* in_sizes, int n_in,
                              void* d_out, int out_size, void* d_ws, size_t ws_size,
                              hipStream_t stream) {
    (void)n_in; (void)d_ws; (void)ws_size;

    const v4f* upd4 = (const v4f*)d_in[0];   // float32 updates [B,H,W,C]
    const v4i* msk4 = (const v4i*)d_in[1];   // int32 mask     [B,H,W,C]
    float*     out  = (float*)d_out;         // float32 output [B,224,224,64]

    const unsigned nOut4 = (unsigned)out_size / 4u;      // 25,690,112 float4s
    const unsigned nIn4  = (unsigned)in_sizes[0] / 4u;   //  6,422,528 quads

    // 1) zero-fill output (must precede scatter; same stream => ordered)
    unsigned zb = (nOut4 + 255u) / 256u;
    unpool_zero_kernel<<<zb, 256, 0, stream>>>((v4f*)out, nOut4);

    // 2) scatter-add
    unsigned sb = (nIn4 + 255u) / 256u;
    unpool_scatter_kernel<<<sb, 256, 0, stream>>>(upd4, msk4, out, nIn4);
}